// DeformableConv1d_46179488366721
// MI455X (gfx1250) — compile-verified
//
#include <hip/hip_runtime.h>
#include <math.h>

typedef float v2f __attribute__((ext_vector_type(2)));
typedef float v8f __attribute__((ext_vector_type(8)));

#define B_ 8
#define L_ 4096
#define C_ 256

// ---------------------------------------------------------------------------
// Kernel 1: both convs (offset + mask) as ONE 16x768 x 768x16 GEMM per wave
// using V_WMMA_F32_16X16X4_F32. A rows 0-2 = w_off, rows 3-5 = w_mask, 6-15 = 0.
// K ordering: kk = t*256 + c  (each K=4 chunk has constant tap t).
//
// A is staged in LDS as pairs: APair[kk>>1][m(0..7)][2], rows 6..7 zero.
// A-frag per lane = one aligned ds_load_b64 at a clamped row (min(row,7)),
// so rows >= 6 read zeros with NO divergent control flow (EXEC untouched).
// ---------------------------------------------------------------------------
__global__ __launch_bounds__(32) void deform_fields_wmma_kernel(
    const float* __restrict__ x,
    const float* __restrict__ w_off, const float* __restrict__ b_off,
    const float* __restrict__ w_mask, const float* __restrict__ b_mask,
    int* __restrict__ FL, float* __restrict__ WF, float* __restrict__ WC)
{
    __shared__ float XLds[18 * C_];     // rows l0-1 .. l0+16, 256 ch (18 KB)
    __shared__ float ALds[384 * 16];    // [kkpair][row0..7][2] (24 KB)

    const int lane = threadIdx.x;
    const int b    = blockIdx.y;
    const int l0   = blockIdx.x * 16;

    // Stage x tile (coalesced: consecutive lanes -> consecutive channels)
    for (int i = lane; i < 18 * C_; i += 32) {
        int r  = i >> 8;
        int cc = i & 255;
        int gl = l0 - 1 + r;
        float v = 0.0f;
        if (gl >= 0 && gl < L_) v = x[((size_t)b * L_ + gl) * C_ + cc];
        XLds[i] = v;
    }
    // Stage combined weight matrix as K-pairs: ALds[p*16 + m*2 + j] = A[m][2p+j]
    for (int i = lane; i < 384 * 16; i += 32) {
        int p  = i >> 4;
        int m  = (i >> 1) & 7;
        int j  = i & 1;
        int kk = (p << 1) + j;
        int t  = kk >> 8;
        int cc = kk & 255;
        float v = 0.0f;
        if (m < 3)      v = w_off [ m      * 768 + cc * 3 + t];
        else if (m < 6) v = w_mask[(m - 3) * 768 + cc * 3 + t];
        ALds[i] = v;
    }
    __syncthreads();

    const int row  = lane & 15;              // M for A-frag / N for B-frag
    const int hi   = lane >> 4;              // K sub-pair select
    const int arow = (row < 7) ? row : 7;    // rows 6..15 -> zero rows, no branch

    v8f acc = {};
    for (int t = 0; t < 3; ++t) {
        const float* xrow  = &XLds[(row + t) * C_ + (hi << 1)];
        const int    kbase = (t << 8) + (hi << 1);   // even
        #pragma unroll 8
        for (int c0 = 0; c0 < C_; c0 += 4) {
            const int ka = kbase + c0;               // even
            v2f a  = *(const v2f*)&ALds[(ka >> 1) * 16 + (arow << 1)]; // ds_load_b64
            v2f bb = *(const v2f*)&xrow[c0];                            // ds_load_b64
            acc = __builtin_amdgcn_wmma_f32_16x16x4_f32(
                false, a, false, bb, (short)0, acc, false, false);
        }
    }

    // D rows 0..5 live in acc[0..5] of lanes 0..15 (N = lane)
    if (hi == 0) {
        const int   l  = l0 + row;
        const float lf = (float)l;
        #pragma unroll
        for (int k = 0; k < 3; ++k) {
            float off = acc[k]     + b_off[k];
            float z   = acc[k + 3] + b_mask[k];
            float mk  = 1.0f / (1.0f + expf(-z));
            float pos = fminf(fmaxf(lf + off, 0.0f), (float)(L_ - 1));
            float flf = floorf(pos);
            float al  = pos - flf;
            int idx = (b * L_ + l) * 3 + k;
            FL[idx] = (int)flf;
            WF[idx] = (1.0f - al) * mk;
            WC[idx] = al * mk;
        }
    }
}

// ---------------------------------------------------------------------------
// Kernel 2: bandwidth-bound gather/combine. Compute phase: lanes span channels
// (coalesced 1KB gathers, L2-resident x). LDS transpose (stride 33,
// conflict-free both directions) so the store phase has lanes spanning l ->
// coalesced 128B writes into the (B,C,L) output layout.
// ---------------------------------------------------------------------------
__global__ __launch_bounds__(256) void deform_sample_kernel(
    const float* __restrict__ x,
    const int* __restrict__ FL, const float* __restrict__ WF,
    const float* __restrict__ WC,
    float* __restrict__ out)
{
    __shared__ float trans[C_ * 33];

    const int tid = threadIdx.x;
    const int b   = blockIdx.y;
    const int l0  = blockIdx.x * 32;
    const int c   = tid;
    const float* xb = x + (size_t)b * L_ * C_;

    for (int j = 0; j < 32; ++j) {
        const int l    = l0 + j;
        const int fidx = (b * L_ + l) * 3;   // block-uniform -> scalar loads
        float acc = 0.0f;
        #pragma unroll
        for (int k = 0; k < 3; ++k) {
            int   fl = FL[fidx + k];
            float wf = WF[fidx + k];
            float wc = WC[fidx + k];
            int   ce = min(fl + 1, L_ - 1);
            acc = fmaf(xb[(size_t)fl * C_ + c], wf, acc);
            acc = fmaf(xb[(size_t)ce * C_ + c], wc, acc);
        }
        trans[c * 33 + j] = acc;
    }
    __syncthreads();

    const int lane = tid & 31;
    const int w    = tid >> 5;
    float* outb = out + (size_t)b * C_ * L_ + l0;
    for (int ci = w; ci < C_; ci += 8) {
        outb[(size_t)ci * L_ + lane] = trans[ci * 33 + lane];  // 128B coalesced
    }
}

extern "C" void kernel_launch(void* const* d_in, const int* in_sizes, int n_in,
                              void* d_out, int out_size, void* d_ws, size_t ws_size,
                              hipStream_t stream) {
    const float* x      = (const float*)d_in[0];
    const float* w_off  = (const float*)d_in[1];
    const float* b_off  = (const float*)d_in[2];
    const float* w_mask = (const float*)d_in[3];
    const float* b_mask = (const float*)d_in[4];
    float* out = (float*)d_out;

    const size_t N3 = (size_t)B_ * 3 * L_;              // 98304 field entries
    int*   FL = (int*)d_ws;
    float* WF = (float*)((char*)d_ws + 1 * N3 * 4);
    float* WC = (float*)((char*)d_ws + 2 * N3 * 4);

    dim3 g1(L_ / 16, B_), blk1(32);
    deform_fields_wmma_kernel<<<g1, blk1, 0, stream>>>(
        x, w_off, b_off, w_mask, b_mask, FL, WF, WC);

    dim3 g2(L_ / 32, B_), blk2(256);
    deform_sample_kernel<<<g2, blk2, 0, stream>>>(x, FL, WF, WC, out);
}